// DivLoss_50560355008552
// MI455X (gfx1250) — compile-verified
//
#include <hip/hip_runtime.h>

// DivLoss scatter-mean: out[n] = mean_{e: src=n, ea0!=0}((x0[dst]-x0[src])/ea0)
//                              + mean_{e: src=n, ea1!=0}((x1[dst]-x1[src])/ea1)
//
// Bound by L2 atomic throughput (~18M atomics) over a 102MB edge stream
// (4.4us HBM floor @ 23.3 TB/s). Strategy: fewest possible HW atomics,
// all landing in one 16B node record; NT hints on the streaming data.

typedef float    v2f __attribute__((ext_vector_type(2)));
typedef _Float16 v2h __attribute__((ext_vector_type(2)));

#define DL_N_NODES 100000
#define DL_N_EDGES 6400000

// Guaranteed no-return HW f32 atomic add at device scope (L2 atomic units).
__device__ __forceinline__ void atomic_add_f32_dev(float* p, float v) {
    asm volatile("global_atomic_add_f32 %0, %1, off scope:SCOPE_DEV"
                 :: "v"(p), "v"(v) : "memory");
}

// Packed 2xf16 atomic add (both mask counters in ONE atomic op).
__device__ __forceinline__ void atomic_pk_add_f16_dev(unsigned* p, unsigned packed) {
    asm volatile("global_atomic_pk_add_f16 %0, %1, off scope:SCOPE_DEV"
                 :: "v"(p), "v"(packed) : "memory");
}

__global__ __launch_bounds__(256) void dl_zero_kernel(float* __restrict__ acc, int n) {
    int i = blockIdx.x * 256 + threadIdx.x;
    if (i < n) acc[i] = 0.0f;
}

// acc record per node (16B, single L2 sector):
//   [0] sum_x (f32)   [1] sum_y (f32)   [2] counts packed 2xf16   [3] pad
__global__ __launch_bounds__(256) void dl_edge_kernel(
    const float* __restrict__ x,      // [N_NODES, 2]
    const float* __restrict__ ea,     // [N_EDGES, 2]
    const int*   __restrict__ ei,     // [2, N_EDGES] (src row, then dst row)
    float*       __restrict__ acc)    // [N_NODES][4]
{
    int e = blockIdx.x * 256 + threadIdx.x;
    if (e >= DL_N_EDGES) return;

    // Streaming, last-use data: non-temporal so the 102MB stream doesn't
    // evict the L2-resident x (800KB) and accumulator (1.6MB) working set.
    int src = __builtin_nontemporal_load(ei + e);
    int dst = __builtin_nontemporal_load(ei + DL_N_EDGES + e);
    v2f eav = __builtin_nontemporal_load(reinterpret_cast<const v2f*>(ea) + e);

    // Random gathers into tiny, hot arrays: regular (RT) cached loads.
    v2f xs = *(reinterpret_cast<const v2f*>(x) + src);
    v2f xd = *(reinterpret_cast<const v2f*>(x) + dst);

    float* a = acc + 4u * (unsigned)src;

    bool mx = (eav.x != 0.0f);
    bool my = (eav.y != 0.0f);

    if (mx) atomic_add_f32_dev(a + 0, (xd.x - xs.x) / eav.x);
    if (my) atomic_add_f32_dev(a + 1, (xd.y - xs.y) / eav.y);

    // One packed-f16 atomic covers both mask counters (0x3C00 == f16 1.0).
    unsigned pk = (mx ? 0x00003C00u : 0u) | (my ? 0x3C000000u : 0u);
    if (pk) atomic_pk_add_f16_dev(reinterpret_cast<unsigned*>(a + 2), pk);
}

__global__ __launch_bounds__(256) void dl_finalize_kernel(
    const float* __restrict__ acc, float* __restrict__ out)
{
    int n = blockIdx.x * 256 + threadIdx.x;
    if (n >= DL_N_NODES) return;
    float sx = acc[4u * n + 0];
    float sy = acc[4u * n + 1];
    v2h  cnt = *reinterpret_cast<const v2h*>(acc + 4u * n + 2);
    float cx = (float)cnt.x;   // exact: f16 holds integers up to 2048
    float cy = (float)cnt.y;
    // cnt>0 implies cnt>=1, so sums/max(cnts,1) == sums/cnts.
    float dx = (cx > 0.0f) ? (sx / cx) : 0.0f;
    float dy = (cy > 0.0f) ? (sy / cy) : 0.0f;
    out[n] = dx + dy;
}

extern "C" void kernel_launch(void* const* d_in, const int* in_sizes, int n_in,
                              void* d_out, int out_size, void* d_ws, size_t ws_size,
                              hipStream_t stream) {
    (void)in_sizes; (void)n_in; (void)out_size; (void)ws_size;

    const float* x  = (const float*)d_in[0];   // [100000, 2] f32
    const float* ea = (const float*)d_in[1];   // [6400000, 2] f32
    const int*   ei = (const int*)d_in[2];     // [2, 6400000] int
    float* out = (float*)d_out;                // [100000] f32
    float* acc = (float*)d_ws;                 // [100000][4] f32 scratch

    const int acc_elems = DL_N_NODES * 4;
    dl_zero_kernel<<<(acc_elems + 255) / 256, 256, 0, stream>>>(acc, acc_elems);

    dl_edge_kernel<<<(DL_N_EDGES + 255) / 256, 256, 0, stream>>>(x, ea, ei, acc);

    dl_finalize_kernel<<<(DL_N_NODES + 255) / 256, 256, 0, stream>>>(acc, out);
}